// AdaptiveConv_67087389163724
// MI455X (gfx1250) — compile-verified
//
#include <hip/hip_runtime.h>
#include <math.h>

// CDNA5 vector types for WMMA f32 16x16x4: A/B = 2 VGPRs (v2f), C/D = 8 VGPRs (v8f)
typedef float v2f __attribute__((ext_vector_type(2)));
typedef float v8f __attribute__((ext_vector_type(8)));

static constexpr int   N_NODES = 100000;
static constexpr int   N_EDGES = 1600000;
static constexpr int   DF      = 50;
static constexpr int   DPAD    = 64;                 // pad rows to 64 f32 = 256B line
static constexpr int   KITER   = 3;
static constexpr float THRESH  = 0.1f / (2.0f * (1.0f - 0.1f)); // gamma*lam
static constexpr int   NTILES  = N_NODES / 16;       // 6250 exactly

// ---------------------------------------------------------------- degrees ----
__global__ void init_deg_kernel(float* outd, float* ind) {
    int i = blockIdx.x * blockDim.x + threadIdx.x;
    if (i < N_NODES) { outd[i] = 1.0f; ind[i] = 1.0f; } // self-loop weight 1
}

__global__ void edge_deg_kernel(const int* __restrict__ src,
                                const int* __restrict__ dst,
                                const float* __restrict__ ew,
                                float* outd, float* ind) {
    int e = blockIdx.x * blockDim.x + threadIdx.x;
    if (e >= N_EDGES) return;
    float w = ew[e];
    atomicAdd(&outd[src[e]], w);   // global_atomic_add_f32 -> L2 atomic units
    atomicAdd(&ind[dst[e]],  w);
}

// in-place deg -> 1/sqrt(deg); deg >= 1 always (self-loop), so no zero guard
__global__ void invsqrt_kernel(float* outd, float* ind, float* wself) {
    int i = blockIdx.x * blockDim.x + threadIdx.x;
    if (i >= N_NODES) return;
    float io = 1.0f / sqrtf(outd[i]);
    float ii = 1.0f / sqrtf(ind[i]);
    outd[i] = io; ind[i] = ii;
    wself[i] = io * ii;            // normalized self-loop weight
}

__global__ void edge_w_kernel(const int* __restrict__ src,
                              const int* __restrict__ dst,
                              const float* __restrict__ ew,
                              const float* __restrict__ iso,
                              const float* __restrict__ isi,
                              float* __restrict__ w) {
    int e = blockIdx.x * blockDim.x + threadIdx.x;
    if (e >= N_EDGES) return;
    w[e] = ew[e] * iso[src[e]] * isi[dst[e]];
}

// ------------------------------------------------------------------- pad -----
__global__ void pad_kernel(const float* __restrict__ feat,
                           float* __restrict__ feat_pad,
                           float* __restrict__ x_pad) {
    int t = blockIdx.x * blockDim.x + threadIdx.x;
    if (t >= N_NODES * DPAD) return;
    int i = t >> 6, k = t & 63;
    float v = (k < DF) ? feat[i * DF + k] : 0.0f;
    feat_pad[t] = v;
    x_pad[t]    = v;
}

// agg = wself[i] * x[i]  (folds the self-loop; no atomics needed for it)
__global__ void agg_init_kernel(const float* __restrict__ x_pad,
                                const float* __restrict__ wself,
                                float* __restrict__ agg) {
    int t = blockIdx.x * blockDim.x + threadIdx.x;
    if (t >= N_NODES * DPAD) return;
    agg[t] = wself[t >> 6] * x_pad[t];
}

// --------------------------------------------------------------- scatter -----
// wave32 per edge: lanes 0..24 each move float2 (2 cols) => 50 f32, one 256B
// row line per wave; 2 f32 L2 atomics per lane. Working set is L2-resident.
__global__ void scatter_kernel(const int* __restrict__ src,
                               const int* __restrict__ dst,
                               const float* __restrict__ w,
                               const float* __restrict__ x_pad,
                               float* __restrict__ agg) {
    int t = blockIdx.x * blockDim.x + threadIdx.x;
    int e = t >> 5;
    int l = t & 31;
    if (e >= N_EDGES) return;
    int   s  = src[e];          // same addr across lanes -> single L1 request
    int   d  = dst[e];
    float we = w[e];
    if (l < 25) {
        const float2 v = *(const float2*)(x_pad + (size_t)s * DPAD + 2 * l);
        float* ap = agg + (size_t)d * DPAD + 2 * l;
        atomicAdd(ap,     v.x * we);
        atomicAdd(ap + 1, v.y * we);
    }
}

// ---------------------------------------------------- update + prox (WMMA) ---
// One wave per 16-row tile. Row sums of diff^2 computed exactly with chained
// V_WMMA_F32_16X16X4_F32 against an all-ones B (D[m][n] = sum_k A[m][k]).
// A layout (32-bit 16x4): lanes 0-15 hold M=lane, K=kb+{0,1}; lanes 16-31 hold
// M=lane-16, K=kb+{2,3}.  D layout: VGPR j -> row j (lanes 0-15) / j+8 (16-31).
__global__ void update_kernel(const float* __restrict__ agg,
                              const float* __restrict__ feat_pad,
                              float* __restrict__ x_pad) {
    int wave = (blockIdx.x * blockDim.x + threadIdx.x) >> 5;
    int lane = threadIdx.x & 31;
    if (wave >= NTILES) return;            // wave-uniform: EXEC all-1s for WMMA

    int row0  = wave * 16;
    int m     = lane & 15;
    int khalf = (lane >> 4) ? 2 : 0;
    const float* ap = agg      + (size_t)(row0 + m) * DPAD;
    const float* fp = feat_pad + (size_t)(row0 + m) * DPAD;

    v8f c = {};
    v2f ones; ones.x = 1.0f; ones.y = 1.0f;

#pragma unroll
    for (int kb = 0; kb < 52; kb += 4) {   // 13 WMMAs cover K=0..51 (pad -> 0)
        int k0 = kb + khalf;
        float d0 = ap[k0]     - fp[k0];
        float d1 = ap[k0 + 1] - fp[k0 + 1];
        v2f a; a.x = d0 * d0; a.y = d1 * d1;
        c = __builtin_amdgcn_wmma_f32_16x16x4_f32(
                false, a, false, ones, (short)0, c, false, false);
    }

    // Phase 2: x = feat + s[r]*(agg - feat).  norm2 of row r lives in c[r&7]
    // on lane 0 (rows 0-7) / lane 16 (rows 8-15), replicated across columns.
#pragma unroll
    for (int r = 0; r < 16; ++r) {
        float n2  = __shfl(c[r & 7], (r < 8) ? 0 : 16, 32);
        float nrm = sqrtf(n2);
        float s   = (nrm > 0.0f) ? fmaxf(nrm - THRESH, 0.0f) / nrm : 0.0f;
        size_t base = (size_t)(row0 + r) * DPAD + 2 * lane;
        float a0 = agg[base],      a1 = agg[base + 1];
        float f0 = feat_pad[base], f1 = feat_pad[base + 1];
        x_pad[base]     = f0 + s * (a0 - f0);   // pad cols: 0 + s*0 stays 0
        x_pad[base + 1] = f1 + s * (a1 - f1);
    }
}

// ----------------------------------------------------------------- unpad -----
__global__ void unpad_kernel(const float* __restrict__ x_pad,
                             float* __restrict__ out) {
    int t = blockIdx.x * blockDim.x + threadIdx.x;
    if (t >= N_NODES * DF) return;
    int i = t / DF, k = t - i * DF;
    out[t] = x_pad[(size_t)i * DPAD + k];
}

// ------------------------------------------------------------------ launch ---
extern "C" void kernel_launch(void* const* d_in, const int* in_sizes, int n_in,
                              void* d_out, int out_size, void* d_ws, size_t ws_size,
                              hipStream_t stream) {
    const float* feat = (const float*)d_in[0];
    const float* ew   = (const float*)d_in[1];
    const int*   src  = (const int*)d_in[2];
    const int*   dst  = (const int*)d_in[3];
    float*       out  = (float*)d_out;

    // workspace carve-up (f32 elements)
    const size_t padElems = (size_t)N_NODES * DPAD;
    float* x_pad    = (float*)d_ws;
    float* feat_pad = x_pad    + padElems;
    float* agg      = feat_pad + padElems;
    float* outd     = agg      + padElems;
    float* ind      = outd     + N_NODES;
    float* wself    = ind      + N_NODES;
    float* w        = wself    + N_NODES;

    const int B = 256;
    const int gN  = (N_NODES + B - 1) / B;
    const int gE  = (N_EDGES + B - 1) / B;
    const int gP  = (N_NODES * DPAD + B - 1) / B;
    const int gS  = (N_EDGES * 32 + B - 1) / B;       // wave per edge
    const int gU  = (NTILES * 32 + B - 1) / B;        // wave per 16-row tile
    const int gO  = (N_NODES * DF + B - 1) / B;

    init_deg_kernel<<<gN, B, 0, stream>>>(outd, ind);
    edge_deg_kernel<<<gE, B, 0, stream>>>(src, dst, ew, outd, ind);
    invsqrt_kernel<<<gN, B, 0, stream>>>(outd, ind, wself);
    edge_w_kernel<<<gE, B, 0, stream>>>(src, dst, ew, outd, ind, w);
    pad_kernel<<<gP, B, 0, stream>>>(feat, feat_pad, x_pad);

    for (int it = 0; it < KITER; ++it) {
        agg_init_kernel<<<gP, B, 0, stream>>>(x_pad, wself, agg);
        scatter_kernel<<<gS, B, 0, stream>>>(src, dst, w, x_pad, agg);
        update_kernel<<<gU, B, 0, stream>>>(agg, feat_pad, x_pad);
    }

    unpad_kernel<<<gO, B, 0, stream>>>(x_pad, out);
}